// DiagLinear_6837587936175
// MI455X (gfx1250) — compile-verified
//
#include <hip/hip_runtime.h>
#include <stdint.h>

// DiagLinear: out[b, c] = x[b, c] * weight[c] + bias[c]
// Memory-bound streaming kernel (256 MiB traffic @ 23.3 TB/s -> ~11.5 us floor).
//
// Strategy:
//  - Each block owns a 1024-column stripe (256 threads x float4) and 32 rows.
//  - weight/bias stripe staged into LDS once per block via CDNA5 async
//    global->LDS b128 copies (ASYNCcnt), read back with ds_load_b128 into
//    registers, amortized over all 32 rows.
//  - x loads and out stores are non-temporal b128 (single-use data; don't
//    thrash the 192MB L2, keep weight/bias lines resident).
//  - Row stride is a template constant for the common 4096-column case so
//    unrolled loads/stores fold into 24-bit immediate offsets (no per-row
//    64-bit VALU address adds).

typedef float v4f __attribute__((ext_vector_type(4)));

#define THREADS        256
#define COLS_PER_BLOCK 1024   // THREADS * 4 floats
#define ROWS_PER_BLOCK 32

// STRIDE4C > 0: compile-time row stride in float4 units; 0: runtime stride.
template <int STRIDE4C>
__global__ __launch_bounds__(THREADS)
void DiagLinear_kernel(const float* __restrict__ x,
                       const float* __restrict__ weight,
                       const float* __restrict__ bias,
                       float* __restrict__ out,
                       int in_size, int batch)
{
    // [0, 4KB): weight stripe, [4KB, 8KB): bias stripe
    __shared__ float lds_wb[2 * COLS_PER_BLOCK];

    const int tid  = threadIdx.x;
    const int col0 = blockIdx.x * COLS_PER_BLOCK;

    // Wave-relative LDS byte offset of the shared array: for LDS generic
    // pointers, addr[31:0] is the LDS offset (ISA aperture mapping). This also
    // keeps the allocation alive and its address escaped.
    const uint32_t lds_base = (uint32_t)(uintptr_t)(void*)lds_wb;
    const uint32_t lds_w    = lds_base + (uint32_t)(tid * 16);
    const uint32_t lds_b    = lds_base + (uint32_t)(COLS_PER_BLOCK * sizeof(float)) +
                              (uint32_t)(tid * 16);

    // ---- Stage weight+bias stripe into LDS with async global->LDS copies.
    // Each thread moves 16B of weight and 16B of bias. EXEC is all-ones here.
    {
        const uint64_t wsrc = (uint64_t)(uintptr_t)(weight + col0 + tid * 4);
        const uint64_t bsrc = (uint64_t)(uintptr_t)(bias   + col0 + tid * 4);
        asm volatile(
            "global_load_async_to_lds_b128 %0, %2, off\n\t"
            "global_load_async_to_lds_b128 %1, %3, off\n\t"
            "s_wait_asynccnt 0"
            :
            : "v"(lds_w), "v"(lds_b), "v"(wsrc), "v"(bsrc)
            : "memory");
    }
    __syncthreads();

    // Read this thread's 4 weights / 4 biases from LDS into registers.
    v4f w4, b4;
    asm volatile(
        "ds_load_b128 %0, %2\n\t"
        "ds_load_b128 %1, %3\n\t"
        "s_wait_dscnt 0"
        : "=&v"(w4), "=&v"(b4)
        : "v"(lds_w), "v"(lds_b)
        : "memory");

    const int    row0    = blockIdx.y * ROWS_PER_BLOCK;
    const size_t stride4 = (STRIDE4C > 0) ? (size_t)STRIDE4C : ((size_t)in_size >> 2);
    const size_t base4   = (size_t)row0 * stride4 + (size_t)((col0 >> 2) + tid);

    const v4f* __restrict__ xv = (const v4f*)x + base4;
    v4f*       __restrict__ ov = (v4f*)out     + base4;

    if (row0 + ROWS_PER_BLOCK <= batch) {
        // Fast path: exact tile, fixed trip count; with STRIDE4C compile-time,
        // each unrolled access is base + immediate offset (r * 16KB).
#pragma unroll 8
        for (int r = 0; r < ROWS_PER_BLOCK; ++r) {
            v4f v = __builtin_nontemporal_load(xv + (size_t)r * stride4);
            v4f o;
            o.x = fmaf(v.x, w4.x, b4.x);
            o.y = fmaf(v.y, w4.y, b4.y);
            o.z = fmaf(v.z, w4.z, b4.z);
            o.w = fmaf(v.w, w4.w, b4.w);
            __builtin_nontemporal_store(o, ov + (size_t)r * stride4);
        }
    } else {
        const int rows = batch - row0;
        for (int r = 0; r < rows; ++r) {
            v4f v = __builtin_nontemporal_load(xv + (size_t)r * stride4);
            v4f o;
            o.x = fmaf(v.x, w4.x, b4.x);
            o.y = fmaf(v.y, w4.y, b4.y);
            o.z = fmaf(v.z, w4.z, b4.z);
            o.w = fmaf(v.w, w4.w, b4.w);
            __builtin_nontemporal_store(o, ov + (size_t)r * stride4);
        }
    }
}

extern "C" void kernel_launch(void* const* d_in, const int* in_sizes, int n_in,
                              void* d_out, int out_size, void* d_ws, size_t ws_size,
                              hipStream_t stream) {
    const float* x      = (const float*)d_in[0];
    const float* weight = (const float*)d_in[1];
    const float* bias   = (const float*)d_in[2];
    float* out          = (float*)d_out;

    const int in_size = in_sizes[1];              // 4096
    const int batch   = in_sizes[0] / in_size;    // 8192

    dim3 block(THREADS);
    dim3 grid(in_size / COLS_PER_BLOCK,
              (batch + ROWS_PER_BLOCK - 1) / ROWS_PER_BLOCK);

    if (in_size == 4096) {
        DiagLinear_kernel<4096 / 4><<<grid, block, 0, stream>>>(
            x, weight, bias, out, in_size, batch);
    } else {
        DiagLinear_kernel<0><<<grid, block, 0, stream>>>(
            x, weight, bias, out, in_size, batch);
    }
}